// GCNModel_3126736192223
// MI455X (gfx1250) — compile-verified
//
#include <hip/hip_runtime.h>

typedef __attribute__((ext_vector_type(2))) float v2f;
typedef __attribute__((ext_vector_type(8))) float v8f;

#define HID 64

// ---------------------------------------------------------------------------
// Padding kernels (layer 1: K=61 -> 64)
// ---------------------------------------------------------------------------
__global__ void pad_x_kernel(const float* __restrict__ x, float* __restrict__ H,
                             int n) {
  int gid = blockIdx.x * blockDim.x + threadIdx.x;
  if (gid >= n * HID) return;
  int i = gid >> 6, f = gid & 63;
  H[gid] = (f < 61) ? x[(size_t)i * 61 + f] : 0.0f;
}

__global__ void pad_w1_kernel(const float* __restrict__ W1,
                              float* __restrict__ Wpad) {
  int gid = blockIdx.x * blockDim.x + threadIdx.x;  // 64*64
  if (gid >= HID * HID) return;
  Wpad[gid] = (gid < 61 * HID) ? W1[gid] : 0.0f;    // row-major (61,64)
}

// ---------------------------------------------------------------------------
// Degree / normalization
// ---------------------------------------------------------------------------
__global__ void deg_init_kernel(float* __restrict__ deg, int n) {
  int i = blockIdx.x * blockDim.x + threadIdx.x;
  if (i < n) deg[i] = 1.0f;  // self-loop contribution
}

__global__ void deg_count_kernel(const int* __restrict__ dst,
                                 float* __restrict__ deg, int ne) {
  int e = blockIdx.x * blockDim.x + threadIdx.x;
  if (e < ne) atomicAdd(&deg[dst[e]], 1.0f);
}

__global__ void dinv_kernel(float* __restrict__ deg, int n) {
  int i = blockIdx.x * blockDim.x + threadIdx.x;
  if (i >= n) return;
  float d = deg[i];
  deg[i] = (d > 0.0f) ? rsqrtf(d) : 0.0f;  // in-place: deg buffer becomes dinv
}

// ---------------------------------------------------------------------------
// WMMA GEMM: OUT(n x 64) = H(n x 64) @ W(64 x 64), full f32 precision.
// One wave computes one 16x16 tile via 16 chained V_WMMA_F32_16X16X4_F32.
// A layout (16x4 f32): lanes 0-15 -> M=lane, VGPR0=K0,VGPR1=K1;
//                      lanes 16-31 -> M=lane-16, VGPR0=K2,VGPR1=K3.
// B layout (4x16):     VGPR0: K=0(+half*2), N=lane&15; VGPR1: K=1(+half*2).
// C/D (16x16 f32):     VGPR r: lanes 0-15 M=r; lanes 16-31 M=r+8.
// ---------------------------------------------------------------------------
__global__ __launch_bounds__(256) void gemm_nk64_wmma(
    const float* __restrict__ H, const float* __restrict__ W,
    float* __restrict__ out, int nrows) {
  int wid  = (int)((blockIdx.x * blockDim.x + threadIdx.x) >> 5);
  int lane = threadIdx.x & 31;
  int ntile = wid & 3;        // 4 col-tiles of 16 over HID=64
  int mtile = wid >> 2;
  if (mtile * 16 >= nrows) return;  // wave-uniform: EXEC stays all-ones
  int row0 = mtile * 16;
  int col0 = ntile * 16;
  int half = lane >> 4;   // 0|1
  int l15  = lane & 15;

  // Preload all 16 B fragments for this col-tile (W is 16KB, L2 resident)
  v2f bfrag[16];
#pragma unroll
  for (int t = 0; t < 16; ++t) {
    int k = t * 4 + half * 2;
    bfrag[t].x = W[(size_t)k * HID + col0 + l15];
    bfrag[t].y = W[(size_t)(k + 1) * HID + col0 + l15];
  }

  v8f acc = {};
  const float* hrow = H + (size_t)(row0 + l15) * HID + half * 2;
#pragma unroll
  for (int t = 0; t < 16; ++t) {
    v2f a;
    a.x = hrow[t * 4 + 0];
    a.y = hrow[t * 4 + 1];
    acc = __builtin_amdgcn_wmma_f32_16x16x4_f32(false, a, false, bfrag[t],
                                                (short)0, acc, false, false);
  }

#pragma unroll
  for (int r = 0; r < 8; ++r) {
    int m = r + half * 8;
    out[(size_t)(row0 + m) * HID + col0 + l15] = acc[r];
  }
}

// ---------------------------------------------------------------------------
// AGG[i][f] = dinv[i]^2 * TMP[i][f] + bias[f]  (self-loop + bias folded in)
// ---------------------------------------------------------------------------
__global__ void agg_init_kernel(const float* __restrict__ TMP,
                                const float* __restrict__ dinv,
                                const float* __restrict__ bias,
                                float* __restrict__ AGG, int n) {
  int gid = blockIdx.x * blockDim.x + threadIdx.x;
  if (gid >= n * HID) return;
  int i = gid >> 6, f = gid & 63;
  float di = dinv[i];
  AGG[gid] = di * di * TMP[gid] + bias[f];
}

// ---------------------------------------------------------------------------
// Edge scatter-add: 16 lanes per edge, float4 gather + 4 f32 atomics.
// TMP (25.6MB) fits in the 192MB L2 -> gathers are L2 hits; atomics resolve
// in L2 atomic units.
// ---------------------------------------------------------------------------
__global__ __launch_bounds__(256) void edge_scatter_kernel(
    const float4* __restrict__ TMP4, const int* __restrict__ src,
    const int* __restrict__ dst, const float* __restrict__ dinv,
    float* __restrict__ AGG, int ne) {
  long long gid = (long long)blockIdx.x * blockDim.x + threadIdx.x;
  int e = (int)(gid >> 4);
  int q = (int)(gid & 15);
  if (e >= ne) return;
  int s = src[e];
  int d = dst[e];
  float nrm = dinv[s] * dinv[d];
  float4 v = TMP4[(size_t)s * 16 + q];
  float* a = AGG + (size_t)d * HID + q * 4;
  atomicAdd(a + 0, nrm * v.x);
  atomicAdd(a + 1, nrm * v.y);
  atomicAdd(a + 2, nrm * v.z);
  atomicAdd(a + 3, nrm * v.w);
}

// ---------------------------------------------------------------------------
// ReLU (vectorized float4)
// ---------------------------------------------------------------------------
__global__ void relu_kernel(const float4* __restrict__ A,
                            float4* __restrict__ H, int n16) {
  int gid = blockIdx.x * blockDim.x + threadIdx.x;
  if (gid >= n16) return;
  float4 v = A[gid];
  v.x = fmaxf(v.x, 0.0f);
  v.y = fmaxf(v.y, 0.0f);
  v.z = fmaxf(v.z, 0.0f);
  v.w = fmaxf(v.w, 0.0f);
  H[gid] = v;
}

// ---------------------------------------------------------------------------
// Head: H = relu(TMP + bh1)
// ---------------------------------------------------------------------------
__global__ void head_bias_relu_kernel(const float* __restrict__ TMP,
                                      const float* __restrict__ bh1,
                                      float* __restrict__ H, int n) {
  int gid = blockIdx.x * blockDim.x + threadIdx.x;
  if (gid >= n * HID) return;
  int f = gid & 63;
  H[gid] = fmaxf(TMP[gid] + bh1[f], 0.0f);
}

// ---------------------------------------------------------------------------
// Final: out[i] = H[i,:] . Wh2 + bh2   (64-wide dot per node; trivial)
// ---------------------------------------------------------------------------
__global__ void head_final_kernel(const float* __restrict__ H,
                                  const float* __restrict__ Wh2,
                                  const float* __restrict__ bh2,
                                  float* __restrict__ out, int n) {
  int i = blockIdx.x * blockDim.x + threadIdx.x;
  if (i >= n) return;
  const float* hr = H + (size_t)i * HID;
  float s = bh2[0];
#pragma unroll
  for (int f = 0; f < HID; ++f) s += hr[f] * Wh2[f];
  out[i] = s;
}

// ---------------------------------------------------------------------------
extern "C" void kernel_launch(void* const* d_in, const int* in_sizes, int n_in,
                              void* d_out, int out_size, void* d_ws,
                              size_t ws_size, hipStream_t stream) {
  const float* x   = (const float*)d_in[0];
  const int*   ei  = (const int*)d_in[1];
  const float* W1  = (const float*)d_in[2];
  const float* b1  = (const float*)d_in[3];
  const float* W2  = (const float*)d_in[4];
  const float* b2  = (const float*)d_in[5];
  const float* b3  = (const float*)d_in[7];
  const float* W3  = (const float*)d_in[6];
  const float* Wh1 = (const float*)d_in[8];
  const float* bh1 = (const float*)d_in[9];
  const float* Wh2 = (const float*)d_in[10];
  const float* bh2 = (const float*)d_in[11];
  float* out = (float*)d_out;

  const int n  = in_sizes[0] / 61;      // 100000
  const int ne = in_sizes[1] / 2;       // 3200000
  const int* src = ei;
  const int* dst = ei + ne;

  // Workspace layout (all 256B aligned)
  const size_t NB = (size_t)n * HID * sizeof(float);  // 25,600,000
  char* ws = (char*)d_ws;
  float* Hbuf = (float*)(ws);               // current activations (n x 64)
  float* TMP  = (float*)(ws + NB);          // h @ W            (n x 64)
  float* AGG  = (float*)(ws + 2 * NB);      // aggregation      (n x 64)
  float* dinv = (float*)(ws + 3 * NB);      // deg -> d^-1/2    (n)
  float* Wpad = (float*)(ws + 3 * NB + 409600);  // padded W1 (64 x 64)

  const int T = 256;
  const int gNF   = (n * HID + T - 1) / T;   // n*64 threads
  const int gN    = (n + T - 1) / T;
  const int gE    = (ne + T - 1) / T;
  const int gE16  = (int)(((long long)ne * 16 + T - 1) / T);
  const int gN16  = (n * 16 + T - 1) / T;
  const int gGemm = ((n / 16) * 4 + 7) / 8;  // waves: (n/16 mtiles)*(4 ntiles), 8/block

  // --- prep: pad x and W1, build dinv -------------------------------------
  pad_x_kernel<<<gNF, T, 0, stream>>>(x, Hbuf, n);
  pad_w1_kernel<<<(HID * HID + T - 1) / T, T, 0, stream>>>(W1, Wpad);
  deg_init_kernel<<<gN, T, 0, stream>>>(dinv, n);
  deg_count_kernel<<<gE, T, 0, stream>>>(dst, dinv, ne);
  dinv_kernel<<<gN, T, 0, stream>>>(dinv, n);

  // --- 3 GCN layers --------------------------------------------------------
  const float* Ws[3] = {Wpad, W2, W3};
  const float* bs[3] = {b1, b2, b3};
  for (int l = 0; l < 3; ++l) {
    gemm_nk64_wmma<<<gGemm, T, 0, stream>>>(Hbuf, Ws[l], TMP, n);
    agg_init_kernel<<<gNF, T, 0, stream>>>(TMP, dinv, bs[l], AGG, n);
    edge_scatter_kernel<<<gE16, T, 0, stream>>>((const float4*)TMP, src, dst,
                                                dinv, AGG, ne);
    relu_kernel<<<gN16, T, 0, stream>>>((const float4*)AGG, (float4*)Hbuf, n * 16);
  }

  // --- MLP head ------------------------------------------------------------
  gemm_nk64_wmma<<<gGemm, T, 0, stream>>>(Hbuf, Wh1, TMP, n);
  head_bias_relu_kernel<<<gNF, T, 0, stream>>>(TMP, bh1, Hbuf, n);
  head_final_kernel<<<gN, T, 0, stream>>>(Hbuf, Wh2, bh2, out, n);
}